// MatchingLayer_57621281243633
// MI455X (gfx1250) — compile-verified
//
#include <hip/hip_runtime.h>
#include <float.h>

#define EPS 1e-8f
#define B_   16
#define S1   128
#define S2   192
#define H_   256
#define FULLW 512

typedef float v2f __attribute__((ext_vector_type(2)));
typedef float v8f __attribute__((ext_vector_type(8)));

__device__ __forceinline__ int f2ord(float f){ int i=__float_as_int(f); return i>=0 ? i : (i ^ 0x7fffffff); }
__device__ __forceinline__ float ord2f(int i){ return __int_as_float(i>=0 ? i : (i ^ 0x7fffffff)); }

__device__ __forceinline__ float wave_sum(float v){
  #pragma unroll
  for (int o=16;o>0;o>>=1) v += __shfl_down(v,o);
  return v;
}

// ---------------- norms: ||q||, ||q*w|| for maxpool weights (W[2+dir]) ----------------
__global__ void norms_kernel(const float* __restrict__ q1, const float* __restrict__ q2,
                             const float* __restrict__ W,
                             float* __restrict__ n1u, float* __restrict__ n2u,
                             float* __restrict__ n1w, float* __restrict__ n2w){
  const int row = blockIdx.x;            // 0..319 : q1 rows then q2 rows
  const int bd  = blockIdx.y;            // 0..31
  const int dir = bd>>4, b = bd&15, dirb = dir*16 + b;
  const int t = threadIdx.x;             // h in [0,256)
  float q;
  if (row < S1) q = q1[(size_t)(b*S1+row)*FULLW + dir*H_ + t];
  else          q = q2[(size_t)(b*S2+(row-S1))*FULLW + dir*H_ + t];
  const float qq = q*q;
  float acc[5];
  acc[0] = qq;
  #pragma unroll
  for (int p=0;p<4;p++){
    float w = W[((2+dir)*4+p)*H_ + t];
    acc[p+1] = qq*w*w;
  }
  __shared__ float red[8][5];
  const int lane = t&31, wid = t>>5;
  #pragma unroll
  for (int k=0;k<5;k++){
    float s = wave_sum(acc[k]);
    if (lane==0) red[wid][k] = s;
  }
  __syncthreads();
  if (t==0){
    float s[5] = {0,0,0,0,0};
    for (int w8=0;w8<8;w8++) for (int k=0;k<5;k++) s[k]+=red[w8][k];
    if (row < S1){
      n1u[dirb*S1+row] = sqrtf(s[0]);
      for (int p=0;p<4;p++) n1w[(dirb*S1+row)*4+p] = fmaxf(sqrtf(s[p+1]), EPS);
    } else {
      int j = row - S1;
      n2u[dirb*S2+j] = sqrtf(s[0]);
      for (int p=0;p<4;p++) n2w[(dirb*S2+j)*4+p] = fmaxf(sqrtf(s[p+1]), EPS);
    }
  }
}

// ---------------- attention: att = (q1 @ q2^T) / where(n1*n2>eps) -----------------
// 128 threads = 4 waves (2x2); each wave computes a 32x32 block (2x2 WMMA tiles).
__global__ void att_kernel(const float* __restrict__ q1, const float* __restrict__ q2,
                           const float* __restrict__ n1u, const float* __restrict__ n2u,
                           float* __restrict__ att){
  const int ntile = blockIdx.x;          // 0..2 -> 64 j cols
  const int mtile = blockIdx.y;          // 0..1 -> 64 i rows
  const int bd = blockIdx.z;
  const int dir = bd>>4, b = bd&15, dirb = dir*16+b;
  const int m0 = mtile*64, n0 = ntile*64;
  const int t = threadIdx.x;
  const int lane = t&31, wid = t>>5;
  const int waveM = wid>>1, waveN = wid&1;
  const int hf = lane>>4, lm = lane&15;

  __shared__ float As[64*36];
  __shared__ float Bs[64*36];

  const float* q1b = q1 + (size_t)(b*S1)*FULLW + dir*H_;
  const float* q2b = q2 + (size_t)(b*S2)*FULLW + dir*H_;

  v8f c00 = {}, c01 = {}, c10 = {}, c11 = {};
  const int ar = t>>3;                   // 0..15
  const int kq = (t&7)*4;
  for (int k0=0;k0<H_;k0+=32){
    #pragma unroll
    for (int rr=0;rr<64;rr+=16){
      float4 av = *(const float4*)(q1b + (size_t)(m0+ar+rr)*FULLW + k0 + kq);
      *(float4*)&As[(ar+rr)*36+kq] = av;
      float4 bv = *(const float4*)(q2b + (size_t)(n0+ar+rr)*FULLW + k0 + kq);
      *(float4*)&Bs[(ar+rr)*36+kq] = bv;
    }
    if (k0+32 < H_){
      __builtin_prefetch(q1b + (size_t)(m0+ar)*FULLW + k0 + 32 + kq, 0, 1);
      __builtin_prefetch(q2b + (size_t)(n0+ar)*FULLW + k0 + 32 + kq, 0, 1);
    }
    __syncthreads();
    #pragma unroll
    for (int kk=0;kk<32;kk+=4){
      v2f a0, a1, b0, b1;
      const int ka = kk + 2*hf;
      a0.x = As[(waveM*32+lm   )*36 + ka];  a0.y = As[(waveM*32+lm   )*36 + ka + 1];
      a1.x = As[(waveM*32+16+lm)*36 + ka];  a1.y = As[(waveM*32+16+lm)*36 + ka + 1];
      b0.x = Bs[(waveN*32+lm   )*36 + ka];  b0.y = Bs[(waveN*32+lm   )*36 + ka + 1];
      b1.x = Bs[(waveN*32+16+lm)*36 + ka];  b1.y = Bs[(waveN*32+16+lm)*36 + ka + 1];
      c00 = __builtin_amdgcn_wmma_f32_16x16x4_f32(false, a0, false, b0, (short)0, c00, false, false);
      c01 = __builtin_amdgcn_wmma_f32_16x16x4_f32(false, a0, false, b1, (short)0, c01, false, false);
      c10 = __builtin_amdgcn_wmma_f32_16x16x4_f32(false, a1, false, b0, (short)0, c10, false, false);
      c11 = __builtin_amdgcn_wmma_f32_16x16x4_f32(false, a1, false, b1, (short)0, c11, false, false);
    }
    __syncthreads();
  }
  const int gj0 = n0 + waveN*32 + lm;
  const float n2v0 = n2u[dirb*S2 + gj0];
  const float n2v1 = n2u[dirb*S2 + gj0 + 16];
  #pragma unroll
  for (int r=0;r<8;r++){
    const int gi0 = m0 + waveM*32 + r + 8*hf;
    const int gi1 = gi0 + 16;
    float nA0 = n1u[dirb*S1+gi0], nA1 = n1u[dirb*S1+gi1];
    float d;
    d = nA0*n2v0; d = d>EPS?d:EPS; att[((size_t)(dirb*S1+gi0))*S2 + gj0   ] = c00[r]/d;
    d = nA0*n2v1; d = d>EPS?d:EPS; att[((size_t)(dirb*S1+gi0))*S2 + gj0+16] = c01[r]/d;
    d = nA1*n2v0; d = d>EPS?d:EPS; att[((size_t)(dirb*S1+gi1))*S2 + gj0   ] = c10[r]/d;
    d = nA1*n2v1; d = d>EPS?d:EPS; att[((size_t)(dirb*S1+gi1))*S2 + gj0+16] = c11[r]/d;
  }
}

// ---------------- max-pool matching: max_j cos(q1*w, q2*w) -----------------
__global__ void maxpool_kernel(const float* __restrict__ q1, const float* __restrict__ q2,
                               const float* __restrict__ W,
                               const float* __restrict__ n1w, const float* __restrict__ n2w,
                               float* __restrict__ out){
  const int mtile = blockIdx.x;          // 0..1 -> 64 rows
  const int p = blockIdx.y;              // 4
  const int bd = blockIdx.z;
  const int dir = bd>>4, b = bd&15, dirb = dir*16+b;
  const int m0 = mtile*64;
  const int t = threadIdx.x;             // 128 threads
  const int lane=t&31, wid=t>>5, waveM=wid>>1, waveN=wid&1, hf=lane>>4, lm=lane&15;

  __shared__ float As[64*36];
  __shared__ float Bs[64*36];
  __shared__ int   redmax[64];
  if (t < 64) redmax[t] = (int)0x80000000;

  const float* q1b = q1 + (size_t)(b*S1)*FULLW + dir*H_;
  const float* q2b = q2 + (size_t)(b*S2)*FULLW + dir*H_;
  const float* wp  = W + ((2+dir)*4+p)*H_;

  float rmax0[8], rmax1[8];
  #pragma unroll
  for (int r=0;r<8;r++){ rmax0[r] = -FLT_MAX; rmax1[r] = -FLT_MAX; }

  const int ar=t>>3, kq=(t&7)*4;
  for (int nc=0;nc<3;nc++){
    const int n0 = nc*64;
    v8f c00 = {}, c01 = {}, c10 = {}, c11 = {};
    for (int k0=0;k0<H_;k0+=32){
      float4 wv = *(const float4*)(wp + k0 + kq);
      #pragma unroll
      for (int rr=0;rr<64;rr+=16){
        float4 av = *(const float4*)(q1b + (size_t)(m0+ar+rr)*FULLW + k0+kq);
        As[(ar+rr)*36+kq+0] = av.x*wv.x*wv.x;
        As[(ar+rr)*36+kq+1] = av.y*wv.y*wv.y;
        As[(ar+rr)*36+kq+2] = av.z*wv.z*wv.z;
        As[(ar+rr)*36+kq+3] = av.w*wv.w*wv.w;
        float4 bv = *(const float4*)(q2b + (size_t)(n0+ar+rr)*FULLW + k0+kq);
        *(float4*)&Bs[(ar+rr)*36+kq] = bv;
      }
      __syncthreads();
      #pragma unroll
      for (int kk=0;kk<32;kk+=4){
        v2f a0, a1, b0, b1;
        const int ka = kk + 2*hf;
        a0.x = As[(waveM*32+lm   )*36 + ka];  a0.y = As[(waveM*32+lm   )*36 + ka + 1];
        a1.x = As[(waveM*32+16+lm)*36 + ka];  a1.y = As[(waveM*32+16+lm)*36 + ka + 1];
        b0.x = Bs[(waveN*32+lm   )*36 + ka];  b0.y = Bs[(waveN*32+lm   )*36 + ka + 1];
        b1.x = Bs[(waveN*32+16+lm)*36 + ka];  b1.y = Bs[(waveN*32+16+lm)*36 + ka + 1];
        c00 = __builtin_amdgcn_wmma_f32_16x16x4_f32(false, a0, false, b0, (short)0, c00, false, false);
        c01 = __builtin_amdgcn_wmma_f32_16x16x4_f32(false, a0, false, b1, (short)0, c01, false, false);
        c10 = __builtin_amdgcn_wmma_f32_16x16x4_f32(false, a1, false, b0, (short)0, c10, false, false);
        c11 = __builtin_amdgcn_wmma_f32_16x16x4_f32(false, a1, false, b1, (short)0, c11, false, false);
      }
      __syncthreads();
    }
    const int gj0 = n0 + waveN*32 + lm;
    const float n2v0 = n2w[(dirb*S2+gj0   )*4+p];
    const float n2v1 = n2w[(dirb*S2+gj0+16)*4+p];
    #pragma unroll
    for (int r=0;r<8;r++){
      const int gi0 = m0 + waveM*32 + r + 8*hf;
      const int gi1 = gi0 + 16;
      const float i0 = 1.f / n1w[(dirb*S1+gi0)*4+p];
      const float i1 = 1.f / n1w[(dirb*S1+gi1)*4+p];
      rmax0[r] = fmaxf(rmax0[r], fmaxf(c00[r]*i0/n2v0, c01[r]*i0/n2v1));
      rmax1[r] = fmaxf(rmax1[r], fmaxf(c10[r]*i1/n2v0, c11[r]*i1/n2v1));
    }
  }
  #pragma unroll
  for (int r=0;r<8;r++){
    atomicMax(&redmax[waveM*32 + r + 8*hf     ], f2ord(rmax0[r]));
    atomicMax(&redmax[waveM*32 + r + 8*hf + 16], f2ord(rmax1[r]));
  }
  __syncthreads();
  if (t < 64){
    int gi = m0 + t;
    out[(size_t)(b*S1+gi)*32 + 8 + dir*4 + p] = ord2f(redmax[t]);
  }
}

// ---------------- mean numerator: meannum = att @ q2  (M=128, N=256, K=192) ----------
__global__ void meannum_kernel(const float* __restrict__ att, const float* __restrict__ q2,
                               float* __restrict__ meannum){
  const int ntile = blockIdx.x;          // 0..3 -> 64 h cols
  const int mtile = blockIdx.y;          // 0..1 -> 64 i rows
  const int bd = blockIdx.z;
  const int dir=bd>>4, b=bd&15, dirb=dir*16+b;
  const int m0=mtile*64, n0=ntile*64;
  const int t=threadIdx.x, lane=t&31, wid=t>>5, waveM=wid>>1, waveN=wid&1, hf=lane>>4, lm=lane&15;
  __shared__ float As[64*36];
  __shared__ float Bs[32*68];
  const float* arow = att + (size_t)(dirb*S1)*S2;
  const float* q2b  = q2  + (size_t)(b*S2)*FULLW + dir*H_;
  v8f c00 = {}, c01 = {}, c10 = {}, c11 = {};
  const int ar=t>>3,  kq=(t&7)*4;        // A: 64 rows x 32 k
  const int br=t>>4,  hq=(t&15)*4;       // B: 32 k-rows x 64 h
  for (int k0=0;k0<S2;k0+=32){
    #pragma unroll
    for (int rr=0;rr<64;rr+=16){
      float4 av = *(const float4*)(arow + (size_t)(m0+ar+rr)*S2 + k0+kq);
      *(float4*)&As[(ar+rr)*36+kq] = av;
    }
    #pragma unroll
    for (int rr=0;rr<32;rr+=8){
      float4 bv = *(const float4*)(q2b + (size_t)(k0+br+rr)*FULLW + n0 + hq);
      *(float4*)&Bs[(br+rr)*68 + hq] = bv;
    }
    __syncthreads();
    #pragma unroll
    for (int kk=0;kk<32;kk+=4){
      v2f a0, a1, b0, b1;
      const int ka = kk + 2*hf;
      a0.x = As[(waveM*32+lm   )*36 + ka];  a0.y = As[(waveM*32+lm   )*36 + ka + 1];
      a1.x = As[(waveM*32+16+lm)*36 + ka];  a1.y = As[(waveM*32+16+lm)*36 + ka + 1];
      b0.x = Bs[(ka  )*68 + waveN*32 + lm]; b0.y = Bs[(ka+1)*68 + waveN*32 + lm];
      b1.x = Bs[(ka  )*68 + waveN*32 + 16 + lm]; b1.y = Bs[(ka+1)*68 + waveN*32 + 16 + lm];
      c00 = __builtin_amdgcn_wmma_f32_16x16x4_f32(false, a0, false, b0, (short)0, c00, false, false);
      c01 = __builtin_amdgcn_wmma_f32_16x16x4_f32(false, a0, false, b1, (short)0, c01, false, false);
      c10 = __builtin_amdgcn_wmma_f32_16x16x4_f32(false, a1, false, b0, (short)0, c10, false, false);
      c11 = __builtin_amdgcn_wmma_f32_16x16x4_f32(false, a1, false, b1, (short)0, c11, false, false);
    }
    __syncthreads();
  }
  const int gh0 = n0 + waveN*32 + lm;
  #pragma unroll
  for (int r=0;r<8;r++){
    const int gi0 = m0 + waveM*32 + r + 8*hf;
    const int gi1 = gi0 + 16;
    meannum[((size_t)(dirb*S1+gi0))*H_ + gh0   ] = c00[r];
    meannum[((size_t)(dirb*S1+gi0))*H_ + gh0+16] = c01[r];
    meannum[((size_t)(dirb*S1+gi1))*H_ + gh0   ] = c10[r];
    meannum[((size_t)(dirb*S1+gi1))*H_ + gh0+16] = c11[r];
  }
}

// ---------------- attsum: row-sum of att ----------------
__global__ void attsum_kernel(const float* __restrict__ att, float* __restrict__ attsum){
  const int i = blockIdx.x, bd = blockIdx.y;
  const int dir=bd>>4, b=bd&15, dirb=dir*16+b;
  const int t = threadIdx.x;             // 32 threads
  const float* row = att + (size_t)(dirb*S1+i)*S2;
  float s = 0.f;
  for (int j=t;j<S2;j+=32) s += row[j];
  s = wave_sum(s);
  if (t==0) attsum[dirb*S1+i] = s;
}

// ---------------- maxatt vector: max_j att[i,j]*q2[j,h] ----------------
__global__ void maxatt_kernel(const float* __restrict__ att, const float* __restrict__ q2,
                              float* __restrict__ maxatt){
  const int i = blockIdx.x, bd = blockIdx.y;
  const int dir=bd>>4, b=bd&15, dirb=dir*16+b;
  const int t = threadIdx.x;             // h
  const float* arow = att + (size_t)(dirb*S1+i)*S2;
  const float* q2c  = q2  + (size_t)(b*S2)*FULLW + dir*H_ + t;
  float m = -FLT_MAX;
  for (int j=0;j<S2;j++) m = fmaxf(m, arow[j]*q2c[(size_t)j*FULLW]);
  maxatt[(size_t)(dirb*S1+i)*H_ + t] = m;
}

// ---------------- full matchings vs a per-(b,i) vector ----------------
__global__ void fullmatch_kernel(const float* __restrict__ q1, const float* __restrict__ q2,
                                 const float* __restrict__ W,
                                 const float* __restrict__ meannum, const float* __restrict__ attsum,
                                 const float* __restrict__ maxatt,
                                 float* __restrict__ out){
  const int i = blockIdx.x, b = blockIdx.y, g = blockIdx.z;  // g in [0,6)
  const int dir = g & 1, kind = g >> 1, dirb = dir*16+b;
  const int t = threadIdx.x;             // h
  const float q1v = q1[(size_t)(b*S1+i)*FULLW + dir*H_ + t];
  float vv; int widx, off;
  if (kind==0){                          // full_fw / full_bw
    const int j = (dir==0) ? (S2-1) : 0;
    vv = q2[(size_t)(b*S2+j)*FULLW + dir*H_ + t];
    widx = dir;     off = dir*4;
  } else if (kind==1){                   // att_match (mean vector)
    float den = attsum[dirb*S1+i]; den = den>EPS ? den : EPS;
    vv = meannum[(size_t)(dirb*S1+i)*H_ + t] / den;
    widx = 4+dir;   off = 16+dir*4;
  } else {                               // maxatt matching
    vv = maxatt[(size_t)(dirb*S1+i)*H_ + t];
    widx = 6+dir;   off = 24+dir*4;
  }
  float acc[12];
  #pragma unroll
  for (int p=0;p<4;p++){
    float w = W[(widx*4+p)*H_ + t];
    float aw = q1v*w, bw = vv*w;
    acc[p*3+0] = aw*bw; acc[p*3+1] = aw*aw; acc[p*3+2] = bw*bw;
  }
  __shared__ float red[8][12];
  const int lane=t&31, wid=t>>5;
  #pragma unroll
  for (int k=0;k<12;k++){
    float s = wave_sum(acc[k]);
    if (lane==0) red[wid][k] = s;
  }
  __syncthreads();
  if (t==0){
    #pragma unroll
    for (int p=0;p<4;p++){
      float dot=0.f, na=0.f, nb=0.f;
      for (int w8=0;w8<8;w8++){ dot+=red[w8][p*3]; na+=red[w8][p*3+1]; nb+=red[w8][p*3+2]; }
      out[(size_t)(b*S1+i)*32 + off + p] =
          dot / (fmaxf(sqrtf(na), EPS) * fmaxf(sqrtf(nb), EPS));
    }
  }
}

extern "C" void kernel_launch(void* const* d_in, const int* in_sizes, int n_in,
                              void* d_out, int out_size, void* d_ws, size_t ws_size,
                              hipStream_t stream){
  (void)in_sizes; (void)n_in; (void)out_size; (void)ws_size;
  const float* q1 = (const float*)d_in[0];   // (16,128,512)
  const float* q2 = (const float*)d_in[1];   // (16,192,512)
  const float* W  = (const float*)d_in[2];   // (8,4,256)
  float* out = (float*)d_out;                // (16,128,32)
  float* ws  = (float*)d_ws;

  float* att     = ws;                       // [2][16][128][192] = 786432
  float* meannum = att     + 786432;         // [2][16][128][256] = 1048576
  float* maxatt  = meannum + 1048576;        // [2][16][128][256] = 1048576
  float* n1u     = maxatt  + 1048576;        // [2][16][128]      = 4096
  float* n2u     = n1u     + 4096;           // [2][16][192]      = 6144
  float* n1w     = n2u     + 6144;           // [2][16][128][4]   = 16384
  float* n2w     = n1w     + 16384;          // [2][16][192][4]   = 24576
  float* attsum  = n2w     + 24576;          // [2][16][128]      = 4096
                                             // total ~11.8 MB

  norms_kernel   <<<dim3(320,32),  256, 0, stream>>>(q1,q2,W,n1u,n2u,n1w,n2w);
  att_kernel     <<<dim3(3,2,32),  128, 0, stream>>>(q1,q2,n1u,n2u,att);
  attsum_kernel  <<<dim3(128,32),   32, 0, stream>>>(att,attsum);
  maxpool_kernel <<<dim3(2,4,32),  128, 0, stream>>>(q1,q2,W,n1w,n2w,out);
  meannum_kernel <<<dim3(4,2,32),  128, 0, stream>>>(att,q2,meannum);
  maxatt_kernel  <<<dim3(128,32),  256, 0, stream>>>(att,q2,maxatt);
  fullmatch_kernel<<<dim3(128,16,6),256, 0, stream>>>(q1,q2,W,meannum,attsum,maxatt,out);
}